// GCN_10282151706868
// MI455X (gfx1250) — compile-verified
//
#include <hip/hip_runtime.h>

// ---------------------------------------------------------------------------
// 2-layer GCN (PyG GCNConv semantics) for MI455X / gfx1250, wave32.
//   deg[v]   = 1 + #{e : col[e]==v};  dinv = rsqrt(deg)
//   h1       = x @ W1                      (fp32 WMMA 16x16x4, B staged in LDS)
//   agg1     = scatter_add(h1[r]*dinv[r]*dinv[c] -> c) + selfloop + b1
//   agg1     = relu(agg1)
//   h2       = agg1 @ W2                   (fp32 WMMA 16x16x4, 40 cols)
//   out      = scatter_add(h2[...]) + selfloop + b2
// All scatter targets (25.6MB / 16MB) are L2-resident (192MB L2), so the
// f32 atomics RMW in L2, not HBM.
// ---------------------------------------------------------------------------

#define GCN_N_NODES 100000
#define GCN_OFF_H   100352                    // dinv rounded up to 512 mult
#define GCN_OFF_AGG (100352 + 6400000)        // after h buffer (100000*64)

typedef float v2f __attribute__((ext_vector_type(2)));
typedef float v8f __attribute__((ext_vector_type(8)));

#if __has_builtin(__builtin_amdgcn_global_load_async_to_lds_b128)
#define GCN_HAVE_ASYNC_LDS 1
// Builtin signature (from hipcc diagnostic): param0 is
//   '__attribute__((vector_size(4*sizeof(int)))) int __device__ *'  (AS1)
typedef int gcn_v4i __attribute__((vector_size(16)));
typedef __attribute__((address_space(1))) gcn_v4i* gcn_gptr_b128;
typedef __attribute__((address_space(3))) gcn_v4i* gcn_lptr_b128;
#else
#define GCN_HAVE_ASYNC_LDS 0
#endif

// device-scope relaxed f32 atomic add -> global_atomic_add_f32 (L2 RMW)
__device__ __forceinline__ void gcn_atomic_add(float* p, float v) {
    __hip_atomic_fetch_add(p, v, __ATOMIC_RELAXED, __HIP_MEMORY_SCOPE_AGENT);
}

// ---------------- utility kernels ----------------

__global__ __launch_bounds__(256) void gcn_fill_f32(float* __restrict__ p, float v, int n) {
    int i = blockIdx.x * blockDim.x + threadIdx.x;
    if (i < n) p[i] = v;
}

// deg[col[e]] += 1.0f  (self-loop "1" pre-seeded by gcn_fill_f32)
__global__ __launch_bounds__(256) void gcn_deg_count(const int* __restrict__ col,
                                                     float* __restrict__ deg, int E) {
    int e = blockIdx.x * blockDim.x + threadIdx.x;
    if (e < E) gcn_atomic_add(&deg[col[e]], 1.0f);
}

// in-place deg -> dinv
__global__ __launch_bounds__(256) void gcn_deg_to_dinv(float* __restrict__ d, int n) {
    int i = blockIdx.x * blockDim.x + threadIdx.x;
    if (i < n) {
        float dv = d[i];
        d[i] = (dv > 0.0f) ? rsqrtf(dv) : 0.0f;
    }
}

// out[v,f] = h[v,f]*dinv[v]^2 + bias[f]   (self-loop message + bias)
__global__ __launch_bounds__(256) void gcn_selfloop_bias(const float* __restrict__ h,
                                                         const float* __restrict__ dinv,
                                                         const float* __restrict__ bias,
                                                         float* __restrict__ out,
                                                         int n_nodes, int F) {
    long long i = (long long)blockIdx.x * blockDim.x + threadIdx.x;
    long long total = (long long)n_nodes * F;
    if (i < total) {
        int v = (int)(i / F);
        int f = (int)(i - (long long)v * F);
        float dv = dinv[v];
        out[i] = h[i] * dv * dv + bias[f];
    }
}

// One wave32 per edge: lanes stride the feature dim (coalesced 128B bursts).
// out[c, f] += h[r, f] * dinv[r] * dinv[c]
template<int F>
__global__ __launch_bounds__(256) void gcn_edge_agg(const float* __restrict__ h,
                                                    const float* __restrict__ dinv,
                                                    const int* __restrict__ row,
                                                    const int* __restrict__ col,
                                                    float* __restrict__ out, int E) {
    int wid  = blockIdx.x * (blockDim.x >> 5) + (threadIdx.x >> 5);
    int lane = threadIdx.x & 31;
    if (wid >= E) return;
    int r = row[wid];
    int c = col[wid];
    float nrm = dinv[r] * dinv[c];
    const float* __restrict__ hs = h + (size_t)r * F;
    float* __restrict__ od = out + (size_t)c * F;
#pragma unroll
    for (int f = lane; f < F; f += 32)
        gcn_atomic_add(&od[f], hs[f] * nrm);
}

__global__ __launch_bounds__(256) void gcn_relu_ip(float* __restrict__ p, long long n) {
    long long i = (long long)blockIdx.x * blockDim.x + threadIdx.x;
    if (i < n) p[i] = fmaxf(p[i], 0.0f);
}

// ---------------- fp32 WMMA GEMM, B staged in LDS ----------------
// D[M x NCOLS] = A[M x K] @ B[K x NCOLS], row-major. One 16-row stripe per
// wave32, NT column tiles of 16 held in registers (A fragment reused NT
// times). B (<=32KB) is staged workgroup-wide into LDS — asynchronously via
// GLOBAL_LOAD_ASYNC_TO_LDS_B128 when the builtin exists — then fragments are
// served by ds_load. Uses V_WMMA_F32_16X16X4_F32.
//   A frag (16x4 f32):  lanes 0-15 hold K={k0,k0+1}, lanes 16-31 K={k0+2,k0+3}
//   B frag (4x16 f32):  mirrored K split, column = lane & 15
//   C/D    (16x16 f32): VGPR i -> row i (lanes 0-15) / row 8+i (lanes 16-31)
template<int NCOLS, int NT, int K>
__global__ __launch_bounds__(256) void gcn_gemm16_wmma_f32(const float* __restrict__ A,
                                                           const float* __restrict__ B,
                                                           float* __restrict__ D,
                                                           int M) {
    __shared__ float ldsB[K * NCOLS];

    // --- stage B workgroup-wide (16B per lane per op) ---
    {
        const int nelem = K * NCOLS;               // multiple of 4 floats
#if GCN_HAVE_ASYNC_LDS
        for (int idx = (int)threadIdx.x * 4; idx < nelem; idx += (int)blockDim.x * 4) {
            __builtin_amdgcn_global_load_async_to_lds_b128(
                (gcn_gptr_b128)(B + idx),
                (gcn_lptr_b128)(ldsB + idx),
                /*offset=*/0, /*cpol=*/0);
        }
#if __has_builtin(__builtin_amdgcn_s_wait_asynccnt)
        __builtin_amdgcn_s_wait_asynccnt(0);
#else
        asm volatile("s_wait_asynccnt 0x0" ::: "memory");
#endif
#else
        for (int idx = (int)threadIdx.x * 4; idx < nelem; idx += (int)blockDim.x * 4)
            *(float4*)(ldsB + idx) = *(const float4*)(B + idx);
#endif
        __syncthreads();
    }

    const int mtiles = M >> 4;
    const int tile = blockIdx.x * (blockDim.x >> 5) + (threadIdx.x >> 5);
    if (tile < mtiles) {                   // wave-uniform: EXEC all-ones inside
        const int lane = threadIdx.x & 31;
        const int g    = lane >> 4;        // half-wave: 0 -> K{0,1}, 1 -> K{2,3}
        const int mn   = lane & 15;        // row (A) / col (B, C, D)
        const int kb   = g * 2;
        const int row0 = tile * 16;

        v8f acc[NT];
#pragma unroll
        for (int t = 0; t < NT; ++t) acc[t] = (v8f){0.f,0.f,0.f,0.f,0.f,0.f,0.f,0.f};

        const float* __restrict__ arow = A + (size_t)(row0 + mn) * K;

        for (int k0 = 0; k0 < K; k0 += 4) {
            v2f a;
            a.x = arow[k0 + kb];
            a.y = arow[k0 + kb + 1];
            const float* brow = ldsB + (k0 + kb) * NCOLS;
#pragma unroll
            for (int t = 0; t < NT; ++t) {
                const int c0 = t * 16 + mn;
                const bool ok = (c0 < NCOLS);     // lane-dependent only in a partial last tile
                v2f b;
                b.x = ok ? brow[c0] : 0.0f;
                b.y = ok ? brow[NCOLS + c0] : 0.0f;
                acc[t] = __builtin_amdgcn_wmma_f32_16x16x4_f32(
                    false, a, false, b, (short)0, acc[t], false, false);
            }
        }

#pragma unroll
        for (int t = 0; t < NT; ++t) {
            const int c = t * 16 + mn;
            if (c < NCOLS) {
#pragma unroll
                for (int i = 0; i < 8; ++i) {
                    const int r = row0 + i + 8 * g;
                    D[(size_t)r * NCOLS + c] = acc[t][i];
                }
            }
        }
    }
}

// ---------------- driver ----------------

extern "C" void kernel_launch(void* const* d_in, const int* in_sizes, int n_in,
                              void* d_out, int out_size, void* d_ws, size_t ws_size,
                              hipStream_t stream) {
    const float* x  = (const float*)d_in[0];   // [100000, 128]
    const int*   ei = (const int*)  d_in[1];   // [2, E] int32
    const float* W1 = (const float*)d_in[2];   // [128, 64]
    const float* b1 = (const float*)d_in[3];   // [64]
    const float* W2 = (const float*)d_in[4];   // [64, 40]
    const float* b2 = (const float*)d_in[5];   // [40]

    const int N = GCN_N_NODES;
    const int E = in_sizes[1] / 2;
    const int* row = ei;
    const int* col = ei + E;

    float* ws   = (float*)d_ws;
    float* dinv = ws;                 // [N]      (deg, then in-place dinv)
    float* h    = ws + GCN_OFF_H;     // [N*64]   (h1, later reused as h2 [N*40])
    float* agg  = ws + GCN_OFF_AGG;   // [N*64]
    float* out  = (float*)d_out;      // [N*40]

    const int TPB = 256;
    const int wavesPerBlk = TPB / 32;

    // --- degrees & normalization ---
    gcn_fill_f32<<<(N + TPB - 1) / TPB, TPB, 0, stream>>>(dinv, 1.0f, N);
    gcn_deg_count<<<(E + TPB - 1) / TPB, TPB, 0, stream>>>(col, dinv, E);
    gcn_deg_to_dinv<<<(N + TPB - 1) / TPB, TPB, 0, stream>>>(dinv, N);

    // --- layer 1: h1 = x @ W1 ---
    {
        int mtiles = N / 16;                                   // 6250
        int blocks = (mtiles + wavesPerBlk - 1) / wavesPerBlk; // 782
        gcn_gemm16_wmma_f32<64, 4, 128><<<blocks, TPB, 0, stream>>>(x, W1, h, N);
    }
    gcn_selfloop_bias<<<((long long)N * 64 + TPB - 1) / TPB, TPB, 0, stream>>>(
        h, dinv, b1, agg, N, 64);
    gcn_edge_agg<64><<<(E + wavesPerBlk - 1) / wavesPerBlk, TPB, 0, stream>>>(
        h, dinv, row, col, agg, E);
    gcn_relu_ip<<<((long long)N * 64 + TPB - 1) / TPB, TPB, 0, stream>>>(
        agg, (long long)N * 64);

    // --- layer 2: h2 = relu(agg1) @ W2 (h buffer reused, width 40) ---
    {
        int mtiles = N / 16;
        int blocks = (mtiles + wavesPerBlk - 1) / wavesPerBlk;
        gcn_gemm16_wmma_f32<40, 3, 64><<<blocks, TPB, 0, stream>>>(agg, W2, h, N);
    }
    gcn_selfloop_bias<<<((long long)N * 40 + TPB - 1) / TPB, TPB, 0, stream>>>(
        h, dinv, b2, out, N, 40);
    gcn_edge_agg<40><<<(E + wavesPerBlk - 1) / wavesPerBlk, TPB, 0, stream>>>(
        h, dinv, row, col, out, E);
}